// GLVQ_45784351375905
// MI455X (gfx1250) — compile-verified
//
#include <hip/hip_runtime.h>
#include <hip/hip_bf16.h>
#include <math.h>

// Problem constants (match reference)
#define NROWS   8192
#define DDIM    1024
#define PROTOS  512
#define PPC     4
#define EPSV    1e-9f

// Tiling
#define ROWS_PER_BLK 32            // rows of x per workgroup
#define KC           32            // K-chunk staged in LDS
#define XSTRIDE      36            // 32 + 4 pad (floats) -> conflict-free frag reads
#define PSTRIDE      36
#define NBLK         (NROWS / ROWS_PER_BLK)   // 256 blocks

typedef float v2f __attribute__((ext_vector_type(2)));
typedef float v8f __attribute__((ext_vector_type(8)));

__device__ __forceinline__ v8f wmma_f32x4(v2f a, v2f b, v8f c) {
  // D = A(16x4,f32) * B(4x16,f32) + C(16x16,f32)
  return __builtin_amdgcn_wmma_f32_16x16x4_f32(
      /*neg_a=*/false, a, /*neg_b=*/false, b,
      /*c_mod=*/(short)0, c, /*reuse_a=*/false, /*reuse_b=*/false);
}

// ---------------------------------------------------------------------------
// Kernel 1: squared L2 norms of x rows (8192) and prototype rows (512).
// One wave per row; 32 floats per lane as float4 streams; shfl reduction.
// ---------------------------------------------------------------------------
__global__ void glvq_sq_kernel(const float* __restrict__ x,
                               const float* __restrict__ protos,
                               float* __restrict__ xsq,
                               float* __restrict__ psq) {
  const int lane = threadIdx.x & 31;
  const int row  = blockIdx.x * 8 + (threadIdx.x >> 5);
  if (row >= NROWS + PROTOS) return;

  const float* src = (row < NROWS) ? (x + (size_t)row * DDIM)
                                   : (protos + (size_t)(row - NROWS) * DDIM);
  float s = 0.0f;
#pragma unroll
  for (int i = 0; i < 8; ++i) {
    float4 v = *(const float4*)(src + ((i * 32 + lane) << 2));
    s += v.x * v.x + v.y * v.y + v.z * v.z + v.w * v.w;
  }
#pragma unroll
  for (int m = 16; m >= 1; m >>= 1) s += __shfl_xor(s, m, 32);
  if (lane == 0) {
    if (row < NROWS) xsq[row] = s;
    else            psq[row - NROWS] = s;
  }
}

// ---------------------------------------------------------------------------
// Kernel 2: main fp32 WMMA GEMM tile + GLVQ epilogue.
// Block = 256 threads (8 waves). Tile = 32 rows x 512 cols (all prototypes).
// Wave w owns cols [64w, 64w+64): 2 row-tiles x 4 col-tiles of 16x16 fp32 acc.
// ---------------------------------------------------------------------------
__global__ void __launch_bounds__(256, 2)
glvq_main_kernel(const float* __restrict__ x,
                 const int*   __restrict__ y,
                 const int*   __restrict__ t_val,
                 const float* __restrict__ protos,
                 const float* __restrict__ xsq,
                 const float* __restrict__ psq,
                 float*       __restrict__ partials) {
  __shared__ float xs_lds[ROWS_PER_BLK * XSTRIDE];   //  4.6 KB
  __shared__ float p_lds[PROTOS * PSTRIDE];          // 73.7 KB
  __shared__ float rmW[8][ROWS_PER_BLK];             // per-wave wrong-class min
  __shared__ float rmC[8][ROWS_PER_BLK];             // per-wave correct-class min
  __shared__ float sigbuf[ROWS_PER_BLK];

  const int tid  = threadIdx.x;
  const int w    = tid >> 5;         // wave 0..7
  const int lane = tid & 31;
  const int ln   = lane & 15;        // column / M-row within half
  const int hi   = lane >> 4;        // half-wave selector
  const int rowBase = blockIdx.x * ROWS_PER_BLK;

  // chunk-staging coordinates
  const int lrow = tid >> 3;         // 0..31
  const int lcol = (tid & 7) << 2;   // 0,4,...,28

  v8f acc[2][4] = {};                // 64 VGPRs of fp32 accumulators

  for (int chunk = 0; chunk < DDIM / KC; ++chunk) {
    const int kc0 = chunk * KC;
    __syncthreads();  // previous chunk's LDS reads done

    // Stage x chunk: 32 rows x 32 cols (1 float4 per thread, coalesced rows)
    {
      float4 v = *(const float4*)(x + (size_t)(rowBase + lrow) * DDIM + kc0 + lcol);
      *(float4*)(&xs_lds[lrow * XSTRIDE + lcol]) = v;
    }
    // Stage proto chunk: 512 rows x 32 cols (16 float4 per thread)
#pragma unroll
    for (int pass = 0; pass < 16; ++pass) {
      const int prow = pass * 32 + lrow;
      float4 v = *(const float4*)(protos + (size_t)prow * DDIM + kc0 + lcol);
      *(float4*)(&p_lds[prow * PSTRIDE + lcol]) = v;
    }
    __syncthreads();

    // 8 K-steps of 4; 8 WMMAs per step (A reused 4x, B reused 2x)
#pragma unroll
    for (int ks = 0; ks < KC; ks += 4) {
      const int kof = ks + 2 * hi;   // fp32 A/B lane layout: halves hold {k,k+1}/{k+2,k+3}
      v2f a0 = *(const v2f*)(&xs_lds[ln * XSTRIDE + kof]);
      v2f a1 = *(const v2f*)(&xs_lds[(16 + ln) * XSTRIDE + kof]);
#pragma unroll
      for (int ct = 0; ct < 4; ++ct) {
        const int n = (w << 6) + (ct << 4) + ln;
        v2f b = *(const v2f*)(&p_lds[n * PSTRIDE + kof]);
        acc[0][ct] = wmma_f32x4(a0, b, acc[0][ct]);
        acc[1][ct] = wmma_f32x4(a1, b, acc[1][ct]);
      }
    }
  }

  // ---- Epilogue: dist = ||x||^2 + ||p||^2 - 2*dot ; masked row minima ----
  const float INF = __builtin_inff();
#pragma unroll
  for (int rt = 0; rt < 2; ++rt) {
#pragma unroll
    for (int r = 0; r < 8; ++r) {
      const int m = rt * 16 + r + 8 * hi;            // local row 0..31 (C layout)
      const int grow = rowBase + m;
      const int yv = y[grow];
      const float xs = xsq[grow];
      float mw = INF, mc = INF;
#pragma unroll
      for (int ct = 0; ct < 4; ++ct) {
        const int n = (w << 6) + (ct << 4) + ln;     // global prototype col
        const float d = xs + psq[n] - 2.0f * acc[rt][ct][r];
        const bool corr = ((n >> 2) == yv);          // PPC == 4
        mw = fminf(mw, corr ? INF : d);
        mc = fminf(mc, corr ? d : INF);
      }
      // min over 16 lanes within each half (columns of this row)
#pragma unroll
      for (int msk = 1; msk <= 8; msk <<= 1) {
        mw = fminf(mw, __shfl_xor(mw, msk, 32));
        mc = fminf(mc, __shfl_xor(mc, msk, 32));
      }
      if (ln == 0) { rmW[w][m] = mw; rmC[w][m] = mc; }
    }
  }
  __syncthreads();

  if (tid < ROWS_PER_BLK) {
    float d1 = INF, d2 = INF;
#pragma unroll
    for (int wv = 0; wv < 8; ++wv) {
      d1 = fminf(d1, rmC[wv][tid]);
      d2 = fminf(d2, rmW[wv][tid]);
    }
    const float mu = (d1 - d2) / (d1 + d2 + EPSV);
    const float alpha = logf(1.0f + (float)t_val[0]);   // ALPHA == 1.0
    sigbuf[tid] = 1.0f / (1.0f + expf(-alpha * mu));
  }
  __syncthreads();

  if (tid == 0) {
    float s = 0.0f;
#pragma unroll
    for (int i = 0; i < ROWS_PER_BLK; ++i) s += sigbuf[i];
    partials[blockIdx.x] = s;
  }
}

// ---------------------------------------------------------------------------
// Kernel 3: reduce per-block partials -> mean
// ---------------------------------------------------------------------------
__global__ void glvq_finalize_kernel(const float* __restrict__ partials,
                                     float* __restrict__ out) {
  __shared__ float buf[NBLK];
  const int tid = threadIdx.x;
  buf[tid] = partials[tid];
  __syncthreads();
#pragma unroll
  for (int s = NBLK / 2; s > 0; s >>= 1) {
    if (tid < s) buf[tid] += buf[tid + s];
    __syncthreads();
  }
  if (tid == 0) out[0] = buf[0] / (float)NROWS;
}

// ---------------------------------------------------------------------------
extern "C" void kernel_launch(void* const* d_in, const int* in_sizes, int n_in,
                              void* d_out, int out_size, void* d_ws, size_t ws_size,
                              hipStream_t stream) {
  const float* x      = (const float*)d_in[0];
  const int*   y      = (const int*)d_in[1];
  const int*   t_val  = (const int*)d_in[2];
  const float* protos = (const float*)d_in[3];

  float* ws       = (float*)d_ws;
  float* xsq      = ws;                    // [8192]
  float* psq      = ws + NROWS;            // [512]
  float* partials = ws + NROWS + PROTOS;   // [256]

  // 1) row norms (one wave per row; 8 waves per block)
  glvq_sq_kernel<<<(NROWS + PROTOS) / 8, 256, 0, stream>>>(x, protos, xsq, psq);

  // 2) WMMA GEMM + GLVQ epilogue
  glvq_main_kernel<<<NBLK, 256, 0, stream>>>(x, y, t_val, protos, xsq, psq, partials);

  // 3) mean
  glvq_finalize_kernel<<<1, NBLK, 0, stream>>>(partials, (float*)d_out);
}